// Sub_Sparse_Clust_Cos_GLM_18451179503637
// MI455X (gfx1250) — compile-verified
//
#include <hip/hip_runtime.h>
#include <math.h>

// ---------------------------------------------------------------------------
// Sub_Sparse_Clust_Cos_GLM for MI455X (gfx1250), wave32, f32 WMMA 16x16x4.
//
// Phases:
//  k_proj : sparsestmax per column of C_syn_raw  -> d_out[40000..88000)
//  k_prep : Ke/Ki = (W_layer1 @ kern_basis) [s][h(pad16)][k],
//           Cw = C_syn * exp(scale) padded to 32 rows, zero syn front-pads
//  k_syn  : syn[b][s][t] = sum_e S[b][t][e]*Cw[s][e]  (WMMA f32 16x16x4,
//           M=sub(pad32), N=time tile 16, K streamed) -- HBM-bound streaming
//  k_conv : h = tanh(causal conv(syn, K, 200 taps) + b1); out = sum expW2*h
//           (WMMA f32: M=hid(pad16), K=200 taps, N=16 timesteps; implicit
//            im2col via shifted loads from zero-padded syn rows)
// ---------------------------------------------------------------------------

typedef float v2f __attribute__((ext_vector_type(2)));
typedef float v8f __attribute__((ext_vector_type(8)));

#define SUB   20
#define HID   10
#define ENO   2000
#define INO   400
#define TAPS  200
#define CB    24
#define TD    20000
#define PAD   256                 // zero front-pad so t-k<0 reads 0
#define ROWLEN (PAD + TD)         // 20256 floats per (b,s) syn row

// workspace layout (float offsets), total 1,825,280 floats (~7.3 MB)
#define CWE_OFF  0u               // 32*2000
#define CWI_OFF  64000u           // 32*400
#define KE_OFF   76800u           // 20*16*200
#define KI_OFF   140800u          // 20*16*200
#define SYNE_OFF 204800u          // 2*20*ROWLEN
#define SYNI_OFF 1015040u         // 2*20*ROWLEN

// D = A(16x4) * B(4x16) + C, f32. Register layouts assumed per CDNA5 ISA
// 7.12.2: A: lane->M (lane&15), VGPR0=K(2*hi), VGPR1=K(2*hi+1);
// B mirrored with lane->N; C/D: VGPR i -> M=i+8*hi, lane->N.
static __device__ inline v8f wmma_f32(v2f a, v2f b, v8f c) {
  return __builtin_amdgcn_wmma_f32_16x16x4_f32(false, a, false, b,
                                               (short)0, c, false, false);
}

// ------------------------- sparsestmax (per column) ------------------------

static __device__ void sparsestmax20(const float* v, float rad_in, float* res) {
  float p[SUB], u_prev[SUB];
  for (int i = 0; i < SUB; i++) { p[i] = v[i]; u_prev[i] = 0.f; }
  float rad_prev = rad_in;
  bool first = true;
  for (int it = 0; it < 12; it++) {
    // ---- sparsemax(p) ----
    float vs[SUB];
    for (int i = 0; i < SUB; i++) vs[i] = p[i];
    for (int i = 1; i < SUB; i++) {            // insertion sort, descending
      float key = vs[i]; int j = i - 1;
      while (j >= 0 && vs[j] < key) { vs[j + 1] = vs[j]; j--; }
      vs[j + 1] = key;
    }
    float css = -1.0f, tau = 0.0f;
    for (int k = 0; k < SUB; k++) {            // last k with vs[k] > css/ind
      css += vs[k];
      float t = css / (float)(k + 1);
      if (vs[k] > t) tau = t;
    }
    float w[SUB], wmax = -1e30f, wmin = 1e30f, cnt = 0.f;
    for (int i = 0; i < SUB; i++) {
      w[i] = fmaxf(p[i] - tau, 0.f);
      wmax = fmaxf(wmax, w[i]); wmin = fminf(wmin, w[i]);
      cnt += (w[i] > 0.f) ? 1.f : 0.f;
    }
    bool c1 = ((wmax - wmin) == 1.0f);
    float inv = 1.0f / cnt;
    float rad_cur;
    if (first) {
      rad_cur = rad_prev;
    } else {
      float du = 0.f;
      for (int i = 0; i < SUB; i++) {
        float ui = (w[i] > 0.f) ? inv : 0.f;
        float d = ui - u_prev[i]; du += d * d;
      }
      rad_cur = sqrtf(fmaxf(rad_prev * rad_prev - du, 1e-12f));
    }
    float dist = 0.f;
    for (int i = 0; i < SUB; i++) {
      float ui = (w[i] > 0.f) ? inv : 0.f;
      float d = w[i] - ui; dist += d * d;
    }
    dist = sqrtf(dist);
    float safe = (dist > 0.f) ? dist : 1.f;
    float pn[SUB], pmin = 1e30f;
    for (int i = 0; i < SUB; i++) {
      float ui = (w[i] > 0.f) ? inv : 0.f;
      pn[i] = rad_cur * (w[i] - ui) / safe + ui;
      pmin = fminf(pmin, pn[i]);
    }
    bool c2 = (dist >= rad_cur);
    bool c3 = (pmin >= 0.f);
    if (c1 || c2) { for (int i = 0; i < SUB; i++) res[i] = w[i]; }
    else          { for (int i = 0; i < SUB; i++) res[i] = fminf(fmaxf(pn[i], 0.f), 1.f); }
    if (c1 || c2 || c3) return;                 // res frozen beyond this point
    for (int i = 0; i < SUB; i++) {
      p[i] = pn[i];
      u_prev[i] = (w[i] > 0.f) ? inv : 0.f;
    }
    rad_prev = rad_cur;
    first = false;
  }
}

__global__ __launch_bounds__(128) void k_proj(const float* __restrict__ Ce_raw,
                                              const float* __restrict__ Ci_raw,
                                              const float* __restrict__ rad,
                                              const int* __restrict__ test,
                                              float* __restrict__ out_Ce,
                                              float* __restrict__ out_Ci) {
  int idx = blockIdx.x * blockDim.x + threadIdx.x;
  const float* raw; float* out; int ncol, col;
  if (idx < ENO)            { raw = Ce_raw; out = out_Ce; ncol = ENO; col = idx; }
  else if (idx < ENO + INO) { raw = Ci_raw; out = out_Ci; ncol = INO; col = idx - ENO; }
  else return;
  float v[SUB];
  for (int s = 0; s < SUB; s++) v[s] = raw[s * ncol + col];
  float r[SUB];
  if (*test != 0) {
    int best = 0; float bv = v[0];
    for (int s = 1; s < SUB; s++) if (v[s] > bv) { bv = v[s]; best = s; }
    for (int s = 0; s < SUB; s++) r[s] = (s == best) ? 1.f : 0.f;
  } else {
    sparsestmax20(v, rad[0], r);
  }
  for (int s = 0; s < SUB; s++) out[s * ncol + col] = r[s];
}

// ------------------------------- weight prep -------------------------------

__global__ __launch_bounds__(256) void k_prep(const float* __restrict__ We,
                                              const float* __restrict__ Wi,
                                              const float* __restrict__ kb,
                                              const float* __restrict__ Ce,
                                              const float* __restrict__ Ci,
                                              const float* __restrict__ Es,
                                              const float* __restrict__ Is,
                                              float* __restrict__ ws) {
  int idx = blockIdx.x * blockDim.x + threadIdx.x;
  if (idx < 128000) {                      // Ke then Ki: [s][h pad16][k]
    const float* W = We; unsigned off = KE_OFF; int local = idx;
    if (local >= 64000) { local -= 64000; W = Wi; off = KI_OFF; }
    int opad = local / TAPS, k = local % TAPS;
    int s = opad >> 4, h = opad & 15;
    float v = 0.f;
    if (h < HID) {
      const float* wrow = W + (s * HID + h) * CB;
      #pragma unroll
      for (int j = 0; j < CB; j++) v += wrow[j] * kb[j * TAPS + k];
    }
    ws[off + local] = v;
  } else if (idx < 204800) {               // Cw_e / Cw_i (pad rows 20..31 = 0)
    int local = idx - 128000;
    if (local < 64000) {
      int row = local / ENO, e = local % ENO;
      ws[CWE_OFF + local] = (row < SUB) ? Ce[row * ENO + e] * expf(Es[e]) : 0.f;
    } else {
      local -= 64000;
      int row = local / INO, e = local % INO;
      ws[CWI_OFF + local] = (row < SUB) ? Ci[row * INO + e] * expf(Is[e]) : 0.f;
    }
  } else if (idx < 225280) {               // zero syn front pads (80 rows)
    int local = idx - 204800;
    int row = local / PAD, off = local % PAD;
    if (row < 40) ws[SYNE_OFF + (size_t)row * ROWLEN + off] = 0.f;
    else          ws[SYNI_OFF + (size_t)(row - 40) * ROWLEN + off] = 0.f;
  }
}

// ------------------ syn GEMM: D[sub(pad32), t16] over K=E ------------------

__global__ __launch_bounds__(128) void k_syn(const float* __restrict__ S,
                                             const float* __restrict__ Cw,
                                             float* __restrict__ syn,
                                             int E, int nTileBlk) {
  int wave = threadIdx.x >> 5;
  int lane = threadIdx.x & 31;
  int lo = lane & 15, hi = lane >> 4;
  int b = blockIdx.x / nTileBlk;
  int tile = (blockIdx.x % nTileBlk) * 4 + wave;
  if (tile >= TD / 16) return;             // wave-uniform exit
  int t0 = tile << 4;
  int th2 = hi << 1;
  const float* srow = S + ((size_t)b * TD + t0 + lo) * (size_t)E;
  const float* a0p = Cw + (size_t)lo * E;          // sub rows 0..15
  const float* a1p = Cw + (size_t)(16 + lo) * E;   // sub rows 16..31 (pad 0)
  v8f c0 = {0.f, 0.f, 0.f, 0.f, 0.f, 0.f, 0.f, 0.f};
  v8f c1 = c0;
  #pragma unroll 4
  for (int e0 = 0; e0 < E; e0 += 4) {
    v2f bb = *(const v2f*)(srow + e0 + th2);       // B[K=th2,K=th2+1; N=lo]
    v2f a0 = *(const v2f*)(a0p + e0 + th2);        // A[M=lo; K=th2,th2+1]
    v2f a1 = *(const v2f*)(a1p + e0 + th2);
    c0 = wmma_f32(a0, bb, c0);
    c1 = wmma_f32(a1, bb, c1);
  }
  // D: VGPR i -> sub = i + 8*hi, lane -> t = t0 + lo
  float* dcol = syn + (size_t)(b * SUB) * ROWLEN + PAD + t0 + lo;
  #pragma unroll
  for (int i = 0; i < 8; i++) {
    int s = i + (hi << 3);
    dcol[(size_t)s * ROWLEN] = c0[i];
  }
  if (hi == 0) {                                    // rows 16..19 of tile 1
    #pragma unroll
    for (int i = 0; i < 4; i++) dcol[(size_t)(16 + i) * ROWLEN] = c1[i];
  }
}

// ------- causal conv (200 taps) + tanh + (s,h)-weighted reduction ----------

__global__ __launch_bounds__(128) void k_conv(const float* __restrict__ synE,
                                              const float* __restrict__ synI,
                                              const float* __restrict__ Ke,
                                              const float* __restrict__ Ki,
                                              const float* __restrict__ W2,
                                              const float* __restrict__ b1,
                                              const float* __restrict__ Vo,
                                              float* __restrict__ out) {
  int wave = threadIdx.x >> 5;
  int lane = threadIdx.x & 31;
  int lo = lane & 15, hi = lane >> 4;
  int tile = blockIdx.x * 4 + wave;        // 0..2499 exactly
  int b = tile / (TD / 16);
  int tt = tile % (TD / 16);
  int t0 = tt << 4;
  int th2 = hi << 1;
  int tl = PAD + t0 + lo;                  // padded index of this lane's t
  float acc = 0.f;
  for (int s = 0; s < SUB; s++) {
    const float* rowE = synE + (size_t)(b * SUB + s) * ROWLEN;
    const float* rowI = synI + (size_t)(b * SUB + s) * ROWLEN;
    const float* keS = Ke + ((size_t)s * 16 + lo) * TAPS;
    const float* kiS = Ki + ((size_t)s * 16 + lo) * TAPS;
    v8f c = {0.f, 0.f, 0.f, 0.f, 0.f, 0.f, 0.f, 0.f};
    #pragma unroll 2
    for (int k0 = 0; k0 < TAPS; k0 += 4) {
      // X[k, t] = syn[t - k]; B VGPR0 = X[k0+th2, t], VGPR1 = X[k0+th2+1, t]
      v2f ae = *(const v2f*)(keS + k0 + th2);
      v2f be; be.x = rowE[tl - k0 - th2]; be.y = rowE[tl - k0 - th2 - 1];
      c = wmma_f32(ae, be, c);
      v2f ai = *(const v2f*)(kiS + k0 + th2);
      v2f bi; bi.x = rowI[tl - k0 - th2]; bi.y = rowI[tl - k0 - th2 - 1];
      c = wmma_f32(ai, bi, c);
    }
    // D: VGPR i -> h = i + 8*hi (valid h < 10), lane -> t
    #pragma unroll
    for (int i = 0; i < 8; i++) {
      int h = i + (hi << 3);
      if (h < HID) {
        int idx = s * HID + h;
        acc += expf(W2[idx]) * tanhf(c[i] + b1[idx]);
      }
    }
  }
  // combine h halves: lanes l and l^16 hold the same timestep t0 + (l&15)
  acc += __shfl_xor(acc, 16, 32);
  if (lane < 16) out[(size_t)b * TD + t0 + lo] = acc + Vo[0];
}

// --------------------------------- launch ----------------------------------

extern "C" void kernel_launch(void* const* d_in, const int* in_sizes, int n_in,
                              void* d_out, int out_size, void* d_ws, size_t ws_size,
                              hipStream_t stream) {
  const float* S_e   = (const float*)d_in[0];
  const float* S_i   = (const float*)d_in[1];
  const float* rad   = (const float*)d_in[2];
  const int*   test  = (const int*)d_in[3];
  const float* Es    = (const float*)d_in[4];
  const float* Is    = (const float*)d_in[5];
  const float* We1   = (const float*)d_in[6];
  const float* Wi1   = (const float*)d_in[7];
  const float* W2    = (const float*)d_in[8];
  const float* b1    = (const float*)d_in[9];
  const float* CeRaw = (const float*)d_in[10];
  const float* CiRaw = (const float*)d_in[11];
  const float* Vo    = (const float*)d_in[12];
  const float* kb    = (const float*)d_in[13];

  float* out       = (float*)d_out;
  float* out_final = out;                 // [2, 20000]
  float* out_Ce    = out + 2 * TD;        // [20, 2000]
  float* out_Ci    = out + 2 * TD + SUB * ENO; // [20, 400]
  float* ws        = (float*)d_ws;        // needs >= 1,825,280 floats

  // 1) sparsestmax projections (writes C_syn outputs directly)
  k_proj<<<(ENO + INO + 127) / 128, 128, 0, stream>>>(CeRaw, CiRaw, rad, test,
                                                      out_Ce, out_Ci);
  // 2) weights: time-domain kernels, scaled C^T, zero syn pads (880*256 exact)
  k_prep<<<880, 256, 0, stream>>>(We1, Wi1, kb, out_Ce, out_Ci, Es, Is, ws);
  // 3) syn GEMMs (HBM-bound stream of S_e / S_i): 2 batches * 313 tile-blocks
  k_syn<<<2 * 313, 128, 0, stream>>>(S_e, ws + CWE_OFF, ws + SYNE_OFF, ENO, 313);
  k_syn<<<2 * 313, 128, 0, stream>>>(S_i, ws + CWI_OFF, ws + SYNI_OFF, INO, 313);
  // 4) conv + tanh + reduce: 2500 wave-tiles / 4 waves per block
  k_conv<<<625, 128, 0, stream>>>(ws + SYNE_OFF, ws + SYNI_OFF,
                                  ws + KE_OFF, ws + KI_OFF,
                                  W2, b1, Vo, out_final);
}